// Loss_2284922601829
// MI455X (gfx1250) — compile-verified
//
#include <hip/hip_runtime.h>

#define BDIM 64
#define CDIM 81
#define NDIM 8732
#define SCALE_XY 10.0f
#define SCALE_WH 5.0f

typedef __attribute__((ext_vector_type(2))) float v2f;
typedef __attribute__((ext_vector_type(8))) float v8f;

// ---------------------------------------------------------------------------
// Kernel 0: zero the per-batch accumulators and the scalar output.
// ---------------------------------------------------------------------------
__global__ void init_kernel(float* acc_loc, float* acc_poscon, int* acc_pos,
                            float* out) {
  int t = threadIdx.x;
  if (t < BDIM) {
    acc_loc[t] = 0.0f;
    acc_poscon[t] = 0.0f;
    acc_pos[t] = 0;
  }
  if (t == 0) out[0] = 0.0f;
}

// ---------------------------------------------------------------------------
// Kernel 1: per-anchor cross entropy (log-softmax over C via WMMA sum-of-exp)
// plus masked smooth-L1 location loss. One wave handles 16 anchors.
//   lanes 0-15  : B-matrix rows K = {c0, c0+1},  column = lane
//   lanes 16-31 : B-matrix rows K = {c0+2, c0+3}, column = lane-16
// A = 16x4 ones  =>  D[i][j] = sum_k B[k][j]  (column sums, f32 exact).
// ---------------------------------------------------------------------------
__global__ __launch_bounds__(256) void ce_loc_kernel(
    const float* __restrict__ ploc, const float* __restrict__ plabel,
    const float* __restrict__ gloc, const int* __restrict__ glabel,
    const float* __restrict__ dboxes, float* __restrict__ ws_con,
    float* acc_loc, float* acc_poscon, int* acc_pos) {
  const int lane = threadIdx.x & 31;
  const int wave = threadIdx.x >> 5;
  const int b = blockIdx.y;
  const int n0 = (blockIdx.x * 8 + wave) * 16;
  if (n0 >= NDIM) return;  // wave-uniform exit, EXEC stays all-ones for WMMA

  const int hb = lane >> 4;   // which half of the wave
  const int col = lane & 15;  // anchor within the 16-wide tile
  const int n = n0 + col;
  const bool valid = (n < NDIM);
  const int nc = valid ? n : (NDIM - 1);  // clamped index for safe loads

  const float* pl = plabel + (size_t)b * CDIM * NDIM;

  // ---- pass 1: running max over classes, split across wave halves --------
  float m = -3.4e38f;
  for (int c = hb; c < CDIM; c += 2)
    m = fmaxf(m, pl[(size_t)c * NDIM + nc]);
  m = fmaxf(m, __shfl_xor(m, 16, 32));  // both halves now hold the per-anchor max

  // ---- pass 2: sum of exp via V_WMMA_F32_16X16X4_F32 ---------------------
  v8f acc = {0.f, 0.f, 0.f, 0.f, 0.f, 0.f, 0.f, 0.f};
  v2f ones;
  ones[0] = 1.0f;
  ones[1] = 1.0f;
  for (int c0 = 0; c0 < 84; c0 += 4) {  // 21 chunks cover C=81 (tail zeroed)
    const int r0 = c0 + 2 * hb;
    v2f bv;
    bv[0] = (r0 < CDIM) ? __expf(pl[(size_t)r0 * NDIM + nc] - m) : 0.0f;
    bv[1] = (r0 + 1 < CDIM) ? __expf(pl[(size_t)(r0 + 1) * NDIM + nc] - m)
                            : 0.0f;
    acc = __builtin_amdgcn_wmma_f32_16x16x4_f32(
        /*neg_a=*/false, ones, /*neg_b=*/false, bv,
        /*c_mod=*/(short)0, acc, /*reuse_a=*/false, /*reuse_b=*/false);
  }
  // every row of D is the column sum; D[0] in each lane = sumexp(anchor col)
  const float lse = m + __logf(acc[0]);

  const int g = glabel[(size_t)b * NDIM + nc];
  const float con = lse - pl[(size_t)g * NDIM + nc];  // -log_softmax at glabel
  const bool pos = (g > 0);

  if (hb == 0 && valid)
    ws_con[(size_t)b * NDIM + n] = pos ? 0.0f : con;  // con_neg for mining

  // ---- smooth-L1 location loss on positive anchors (low half only) -------
  float locs = 0.0f, poscon = 0.0f;
  int posn = 0;
  if (hb == 0 && valid && pos) {
    const float* pp = ploc + (size_t)b * 4 * NDIM;
    const float* gg = gloc + (size_t)b * 4 * NDIM;
    const float dx = dboxes[0 * NDIM + n], dy = dboxes[1 * NDIM + n];
    const float dw = dboxes[2 * NDIM + n], dh = dboxes[3 * NDIM + n];
    const float v0 = SCALE_XY * (gg[0 * NDIM + n] - dx) / dw;
    const float v1 = SCALE_XY * (gg[1 * NDIM + n] - dy) / dh;
    const float v2 = SCALE_WH * __logf(gg[2 * NDIM + n] / dw);
    const float v3 = SCALE_WH * __logf(gg[3 * NDIM + n] / dh);
    const float d[4] = {pp[0 * NDIM + n] - v0, pp[1 * NDIM + n] - v1,
                        pp[2 * NDIM + n] - v2, pp[3 * NDIM + n] - v3};
#pragma unroll
    for (int i = 0; i < 4; ++i) {
      const float ad = fabsf(d[i]);
      locs += (ad < 1.0f) ? 0.5f * d[i] * d[i] : (ad - 0.5f);
    }
    poscon = con;
    posn = 1;
  }

  // ---- wave reduction + one atomic per wave ------------------------------
  for (int off = 16; off; off >>= 1) {
    locs += __shfl_xor(locs, off, 32);
    poscon += __shfl_xor(poscon, off, 32);
    posn += __shfl_xor(posn, off, 32);
  }
  if (lane == 0) {
    atomicAdd(&acc_loc[b], locs);
    atomicAdd(&acc_poscon[b], poscon);
    atomicAdd(&acc_pos[b], posn);
  }
}

// ---------------------------------------------------------------------------
// Kernel 2: hard-negative mining per batch row. con_neg >= 0, so float order
// equals uint order of bit patterns: 32-step binary search for the k-th
// largest value T, then neg_sum = sum(x > T) + (k - count_gt) * T (exact
// under ties). Finishes the loss and atomically adds the batch contribution.
// ---------------------------------------------------------------------------
__global__ __launch_bounds__(256) void hnm_kernel(
    const float* __restrict__ ws_con, const float* __restrict__ acc_loc,
    const float* __restrict__ acc_poscon, const int* __restrict__ acc_pos,
    float* out) {
  __shared__ int s_cnt[256];
  __shared__ float s_sum[256];
  const int b = blockIdx.x;
  const int tid = threadIdx.x;
  const float* cn = ws_con + (size_t)b * NDIM;
  const int pos = acc_pos[b];
  const int k = min(3 * pos, NDIM);

  float neg_sum = 0.0f;
  if (k > 0) {
    unsigned v = 0u;
    for (int bit = 31; bit >= 0; --bit) {
      const unsigned cand = v | (1u << bit);
      const float t = __uint_as_float(cand);
      int c = 0;
      for (int i = tid; i < NDIM; i += 256) c += (cn[i] >= t) ? 1 : 0;
      s_cnt[tid] = c;
      __syncthreads();
      for (int s = 128; s; s >>= 1) {
        if (tid < s) s_cnt[tid] += s_cnt[tid + s];
        __syncthreads();
      }
      if (s_cnt[0] >= k) v = cand;  // uniform decision across the block
      __syncthreads();
    }
    const float T = __uint_as_float(v);
    int c = 0;
    float s = 0.0f;
    for (int i = tid; i < NDIM; i += 256) {
      const float x = cn[i];
      if (x > T) {
        c++;
        s += x;
      }
    }
    s_cnt[tid] = c;
    s_sum[tid] = s;
    __syncthreads();
    for (int st = 128; st; st >>= 1) {
      if (tid < st) {
        s_cnt[tid] += s_cnt[tid + st];
        s_sum[tid] += s_sum[tid + st];
      }
      __syncthreads();
    }
    neg_sum = s_sum[0] + (float)(k - s_cnt[0]) * T;
  }

  if (tid == 0) {
    const float total = acc_loc[b] + acc_poscon[b] + neg_sum;
    const float contrib =
        (pos > 0) ? total / (float)pos / (float)BDIM : 0.0f;
    atomicAdd(out, contrib);
  }
}

// ---------------------------------------------------------------------------
extern "C" void kernel_launch(void* const* d_in, const int* in_sizes, int n_in,
                              void* d_out, int out_size, void* d_ws,
                              size_t ws_size, hipStream_t stream) {
  const float* ploc = (const float*)d_in[0];    // [B,4,N]
  const float* plabel = (const float*)d_in[1];  // [B,C,N]
  const float* gloc = (const float*)d_in[2];    // [B,4,N]
  const int* glabel = (const int*)d_in[3];      // [B,N]
  const float* dboxes = (const float*)d_in[4];  // [1,4,N]
  float* out = (float*)d_out;                   // scalar

  float* ws_con = (float*)d_ws;                 // B*N con_neg values
  float* acc_loc = ws_con + (size_t)BDIM * NDIM;
  float* acc_poscon = acc_loc + BDIM;
  int* acc_pos = (int*)(acc_poscon + BDIM);

  init_kernel<<<1, 128, 0, stream>>>(acc_loc, acc_poscon, acc_pos, out);

  dim3 grid((NDIM + 127) / 128, BDIM);  // 128 anchors per 8-wave block
  ce_loc_kernel<<<grid, 256, 0, stream>>>(ploc, plabel, gloc, glabel, dboxes,
                                          ws_con, acc_loc, acc_poscon,
                                          acc_pos);

  hnm_kernel<<<BDIM, 256, 0, stream>>>(ws_con, acc_loc, acc_poscon, acc_pos,
                                       out);
}